// ATSPEdgeEmbedding_82300163326182
// MI455X (gfx1250) — compile-verified
//
#include <hip/hip_runtime.h>

// ---------------- problem constants (from reference) ----------------
constexpr int    B_   = 32;
constexpr int    N_   = 1000;
constexpr int    D_   = 128;
constexpr int    K_   = 50;
constexpr long   E_   = (long)B_ * N_ * K_;           // 1,600,000 edges
constexpr size_t SRC_OFF = 0;
constexpr size_t DST_OFF = (size_t)E_;                // 1,600,000
constexpr size_t EMB_OFF = (size_t)2 * E_;            // 3,200,000
constexpr size_t X_OFF   = EMB_OFF + (size_t)E_ * D_; // 208,000,000
constexpr float  FLT_BIG = 3.402823466e+38f;

typedef __attribute__((ext_vector_type(2))) float v2f;
typedef __attribute__((ext_vector_type(8))) float v8f;

// ------------------------------------------------------------------
// Kernel 1: per-row top-50-smallest (value+index, ascending), wave32/row.
// Writes src/dst edge indices (as exact f32 integers) and stages each
// edge's cost value at edge_emb[e*D] for kernel 2 to consume.
// ------------------------------------------------------------------
__global__ __launch_bounds__(256) void topk50_kernel(
    const float* __restrict__ cost, float* __restrict__ out) {
  const int lane = threadIdx.x & 31;
  const int wid  = threadIdx.x >> 5;
  const int r    = blockIdx.x * 8 + wid;          // row id 0..31999 (= n + b*N)
  const float* rowp = cost + (size_t)r * N_;

  // Each lane owns elements lane, lane+32, ..., padded with +MAX.
  float v[32];
#pragma unroll
  for (int j = 0; j < 32; ++j) {
    const int idx = lane + j * 32;
    v[j] = (idx < N_) ? rowp[idx] : FLT_BIG;      // coalesced 128B per step
  }

  const int    bN    = (r / N_) * N_;             // batch offset b*N
  const size_t ebase = (size_t)r * K_;

  for (int t = 0; t < K_; ++t) {
    // local argmin over this lane's 32 elements (fully unrolled)
    float lmin = v[0];
    int   li   = lane;
#pragma unroll
    for (int j = 1; j < 32; ++j) {
      const int  idx = lane + j * 32;
      const bool c   = v[j] < lmin;
      lmin = c ? v[j] : lmin;
      li   = c ? idx  : li;
    }
    // wave32 argmin, tie-break on lowest index (matches lax.top_k)
#pragma unroll
    for (int off = 16; off > 0; off >>= 1) {
      const float ov = __shfl_xor(lmin, off, 32);
      const int   oi = __shfl_xor(li,   off, 32);
      const bool take = (ov < lmin) || (ov == lmin && oi < li);
      lmin = take ? ov : lmin;
      li   = take ? oi : li;
    }
    if (lane == 0) {
      const size_t e = ebase + t;
      out[SRC_OFF + e]            = (float)r;          // n + b*N == r, exact
      out[DST_OFF + e]            = (float)(li + bN);  // < 2^24, exact
      out[EMB_OFF + e * (size_t)D_] = lmin;            // stage value for kernel 2
    }
    // invalidate extracted element (exactly one lane matches)
    const bool mine = (li & 31) == lane;
    const int  jm   = li >> 5;
#pragma unroll
    for (int j = 0; j < 32; ++j)
      v[j] = (mine && jm == j) ? FLT_BIG : v[j];
  }
}

// ------------------------------------------------------------------
// Kernel 2: edge_emb[e,d] = vals[e]*W[d] + b[d] via V_WMMA_F32_16X16X4_F32.
// One wave per 16-edge group; 8 N-tiles cover D=128.
//   A (16x4 f32): lanes 0-15 VGPR0 = vals[M=lane] at K=0, everything else 0.
//   B (4x16 f32): lanes 0-15 VGPR0 = W[col]       at K=0, everything else 0.
//   C (16x16):    bias broadcast down each column.
// ------------------------------------------------------------------
__global__ __launch_bounds__(256) void edge_emb_wmma_kernel(
    const float* __restrict__ W, const float* __restrict__ bias,
    float* __restrict__ out) {
  const int  lane = threadIdx.x & 31;
  const int  wid  = threadIdx.x >> 5;
  const long g    = (long)blockIdx.x * 8 + wid;   // 0..99,999 edge groups
  const size_t e0 = (size_t)g * 16;
  const int  half = lane >> 4;                    // 0: lanes 0-15, 1: 16-31
  const int  l16  = lane & 15;

  // staged values (read before this wave overwrites column 0 below)
  const float aval = out[EMB_OFF + (e0 + l16) * (size_t)D_];
  v2f a;
  a.x = half ? 0.0f : aval;   // K=0 (lanes 0-15) / K=2 (lanes 16-31, zero)
  a.y = 0.0f;                 // K=1 / K=3

#pragma unroll
  for (int tn = 0; tn < 8; ++tn) {
    const int   col = tn * 16 + l16;
    const float wv  = W[col];
    const float bv  = bias[col];
    v2f bm;
    bm.x = half ? 0.0f : wv;
    bm.y = 0.0f;
    v8f c;
#pragma unroll
    for (int q = 0; q < 8; ++q) c[q] = bv;        // bias broadcast over rows
    // D = A*B + C : full-precision f32 rank-1 update on the matrix pipe
    v8f d = __builtin_amdgcn_wmma_f32_16x16x4_f32(
        false, a, false, bm, (short)0, c, false, false);
    const size_t rbase = e0 + (size_t)half * 8;   // rows: lanes0-15 -> 0..7
#pragma unroll
    for (int q = 0; q < 8; ++q)                   // 64B contiguous across lanes
      out[EMB_OFF + (rbase + q) * (size_t)D_ + col] = d[q];
  }
}

// ------------------------------------------------------------------
// Kernel 3: x = init_embedding (float4 copy, 16 MB)
// ------------------------------------------------------------------
__global__ __launch_bounds__(256) void copy_x_kernel(
    const float4* __restrict__ src, float4* __restrict__ dst) {
  const size_t i = (size_t)blockIdx.x * blockDim.x + threadIdx.x;
  dst[i] = src[i];
}

extern "C" void kernel_launch(void* const* d_in, const int* in_sizes, int n_in,
                              void* d_out, int out_size, void* d_ws, size_t ws_size,
                              hipStream_t stream) {
  (void)in_sizes; (void)n_in; (void)out_size; (void)d_ws; (void)ws_size;
  const float* cost = (const float*)d_in[0];   // [32,1000,1000] f32
  const float* x    = (const float*)d_in[1];   // [32,1000,128]  f32
  const float* W    = (const float*)d_in[2];   // [128,1]        f32
  const float* bias = (const float*)d_in[3];   // [128]          f32
  float* out = (float*)d_out;

  // 32,000 rows, 8 waves/block -> 4000 blocks
  topk50_kernel<<<4000, 256, 0, stream>>>(cost, out);
  // 100,000 16-edge groups, 8 waves/block -> 12,500 blocks
  edge_emb_wmma_kernel<<<12500, 256, 0, stream>>>(W, bias, out);
  // 1,024,000 float4s -> 4000 blocks
  copy_x_kernel<<<4000, 256, 0, stream>>>((const float4*)x,
                                          (float4*)(out + X_OFF));
}